// GAT_3925600108646
// MI455X (gfx1250) — compile-verified
//
#include <hip/hip_runtime.h>
#include <hip/hip_bf16.h>
#include <stdint.h>

typedef float v2f __attribute__((ext_vector_type(2)));
typedef float v8f __attribute__((ext_vector_type(8)));
typedef int   v4i __attribute__((ext_vector_type(4)));

#define HEADS 8
#define CH 10
#define DOUT 80   // HEADS*CH
#define NEG_SLOPE 0.2f

#if defined(__gfx1250__)
#define HAS_ASYNC_LDS (__has_builtin(__builtin_amdgcn_global_load_async_to_lds_b128) && \
                       __has_builtin(__builtin_amdgcn_s_wait_asynccnt))
#else
#define HAS_ASYNC_LDS 0
#endif

// ---------- helpers ----------
__device__ __forceinline__ unsigned f2ord(float f) {
    unsigned u = __float_as_uint(f);
    return (u & 0x80000000u) ? ~u : (u | 0x80000000u);
}
__device__ __forceinline__ float ord2f(unsigned u) {
    return (u & 0x80000000u) ? __uint_as_float(u & 0x7fffffffu)
                             : __uint_as_float(~u);
}

// ---------- fill ----------
__global__ void fill_u32(unsigned* __restrict__ p, unsigned v, long long n) {
    long long t = (long long)blockIdx.x * blockDim.x + threadIdx.x;
    if (t < n) p[t] = v;
}

// ---------- fused GEMM: Yl = X@Wl, Yr = X@Wr  (X:[N,K], W:[K,80]) ----------
// One wave32 per 16-row strip; computes all 5 column tiles for BOTH weight
// matrices (10 x v8f accumulators). Wl/Wr staged in LDS per block via
// GLOBAL_LOAD_ASYNC_TO_LDS_B128 when available.
// KT > 0: compile-time K (full unroll); KT == 0: runtime K fallback.
template<int KT>
__global__ void gemm2_wmma(const float* __restrict__ X,
                           const float* __restrict__ Wlg,
                           const float* __restrict__ Wrg,
                           float* __restrict__ Yl,
                           float* __restrict__ Yr,
                           int nrows, int Krt) {
    const int K = (KT > 0) ? KT : Krt;
    extern __shared__ float lds[];           // [2][K][DOUT]
    float* wl = lds;
    float* wr = lds + (size_t)K * DOUT;

    // --- stage both weight matrices into LDS ---
#if HAS_ASYNC_LDS
    {
        typedef __attribute__((address_space(1))) v4i gv4i;
        typedef __attribute__((address_space(3))) v4i sv4i;
        // 16 bytes per lane per step; K*DOUT is a multiple of 4 floats
        for (int i = threadIdx.x * 4; i < K * DOUT; i += blockDim.x * 4) {
            __builtin_amdgcn_global_load_async_to_lds_b128(
                (gv4i*)(uintptr_t)(Wlg + i), (sv4i*)(uintptr_t)(wl + i), 0, 0);
            __builtin_amdgcn_global_load_async_to_lds_b128(
                (gv4i*)(uintptr_t)(Wrg + i), (sv4i*)(uintptr_t)(wr + i), 0, 0);
        }
        __builtin_amdgcn_s_wait_asynccnt(0);
    }
#else
    for (int i = threadIdx.x; i < K * DOUT; i += blockDim.x) {
        wl[i] = Wlg[i];
        wr[i] = Wrg[i];
    }
#endif
    __syncthreads();

    const int lane = threadIdx.x & 31;
    const int wave = blockIdx.x * (blockDim.x >> 5) + (threadIdx.x >> 5);
    if (wave * 16 >= nrows) return;

    const int l15  = lane & 15;
    const int half = lane >> 4;              // 0 or 1
    const int khi  = half * 2;               // lanes 16-31 hold K+2,K+3
    const long long row = (long long)wave * 16 + l15;

    v8f accl[5] = {};
    v8f accr[5] = {};

#if __has_builtin(__builtin_amdgcn_wmma_f32_16x16x4_f32)
    const float* xrow = X + row * K;
    #pragma unroll
    for (int k = 0; k < K; k += 4) {
        v2f a;
        a.x = xrow[k + khi];                 // one b64 load per lane per step
        a.y = xrow[k + khi + 1];
        const float* wl0 = wl + (k + khi) * DOUT;
        const float* wr0 = wr + (k + khi) * DOUT;
        #pragma unroll
        for (int tn = 0; tn < 5; ++tn) {
            const int col = tn * 16 + l15;
            v2f bl, br;
            bl.x = wl0[col];       bl.y = wl0[DOUT + col];
            br.x = wr0[col];       br.y = wr0[DOUT + col];
            accl[tn] = __builtin_amdgcn_wmma_f32_16x16x4_f32(
                           false, a, false, bl, (short)0, accl[tn], false, false);
            accr[tn] = __builtin_amdgcn_wmma_f32_16x16x4_f32(
                           false, a, false, br, (short)0, accr[tn], false, false);
        }
    }
#else
    // scalar fallback with identical C lane mapping
    const int mrow_f = wave * 16 + half * 8;
    #pragma unroll
    for (int tn = 0; tn < 5; ++tn) {
        const int col = tn * 16 + l15;
        for (int i = 0; i < 8; ++i) {
            float al = 0.f, ar = 0.f;
            const float* xr_ = X + (long long)(mrow_f + i) * K;
            for (int k = 0; k < K; ++k) {
                al = fmaf(xr_[k], wl[k * DOUT + col], al);
                ar = fmaf(xr_[k], wr[k * DOUT + col], ar);
            }
            accl[tn][i] = al;
            accr[tn][i] = ar;
        }
    }
#endif

    // C/D layout: VGPR i -> M = i + 8*half, N = lane&15
    const int mrow = wave * 16 + half * 8;
    #pragma unroll
    for (int tn = 0; tn < 5; ++tn) {
        const int col = tn * 16 + l15;
        #pragma unroll
        for (int i = 0; i < 8; ++i) {
            Yl[(long long)(mrow + i) * DOUT + col] = accl[tn][i];
            Yr[(long long)(mrow + i) * DOUT + col] = accr[tn][i];
        }
    }
}

// ---------- edge pass 1: e = att . leaky_relu(xl[src]+xr[dst]); segment max ----------
__global__ void edge_e_kernel(const float* __restrict__ xl,
                              const float* __restrict__ xr,
                              const int* __restrict__ srcA,
                              const int* __restrict__ dstA,
                              const float* __restrict__ att,
                              float* __restrict__ e,
                              unsigned* __restrict__ nmax,
                              int E, int N) {
    long long t = (long long)blockIdx.x * blockDim.x + threadIdx.x;
    long long total = (long long)(E + N) * HEADS;
    if (t >= total) return;
    const int h = (int)(t & (HEADS - 1));
    const long long eid = t >> 3;
    int s, d;
    if (eid < E) { s = srcA[eid]; d = dstA[eid]; }
    else         { s = d = (int)(eid - E); }

    const float* pl = xl + (long long)s * DOUT + h * CH;
    const float* pr = xr + (long long)d * DOUT + h * CH;
    float acc = 0.f;
    #pragma unroll
    for (int c = 0; c < CH; ++c) {
        float g = pl[c] + pr[c];
        g = (g > 0.f) ? g : NEG_SLOPE * g;
        acc = fmaf(att[h * CH + c], g, acc);
    }
    e[eid * HEADS + h] = acc;
    atomicMax(&nmax[(long long)d * HEADS + h], f2ord(acc));
}

// ---------- edge pass 2: p = exp(e - max); segment sum ----------
__global__ void edge_p_kernel(float* __restrict__ e,
                              const unsigned* __restrict__ nmax,
                              float* __restrict__ denom,
                              const int* __restrict__ dstA,
                              int E, int N) {
    long long t = (long long)blockIdx.x * blockDim.x + threadIdx.x;
    long long total = (long long)(E + N) * HEADS;
    if (t >= total) return;
    const int h = (int)(t & (HEADS - 1));
    const long long eid = t >> 3;
    int d = (eid < E) ? dstA[eid] : (int)(eid - E);
    float m = ord2f(nmax[(long long)d * HEADS + h]);
    float p = __expf(e[eid * HEADS + h] - m);
    e[eid * HEADS + h] = p;
    atomicAdd(&denom[(long long)d * HEADS + h], p);
}

// ---------- edge pass 3: scatter alpha * xl[src] into acc[dst] ----------
__global__ void edge_scatter_kernel(const float* __restrict__ xl,
                                    const float* __restrict__ p,
                                    const float* __restrict__ denom,
                                    const int* __restrict__ srcA,
                                    const int* __restrict__ dstA,
                                    float* __restrict__ acc,
                                    int E, int N) {
    long long t = (long long)blockIdx.x * blockDim.x + threadIdx.x;
    long long total = (long long)(E + N) * DOUT;
    if (t >= total) return;
    const int j = (int)(t % DOUT);
    const long long eid = t / DOUT;
    int s, d;
    if (eid < E) { s = srcA[eid]; d = dstA[eid]; }
    else         { s = d = (int)(eid - E); }
    const int h = j / CH;
    float alpha = p[eid * HEADS + h] / denom[(long long)d * HEADS + h];
    atomicAdd(&acc[(long long)d * DOUT + j], alpha * xl[(long long)s * DOUT + j]);
}

// ---------- finalize: xnext = elu(acc + bias) ----------
__global__ void finalize_kernel(const float* __restrict__ acc,
                                const float* __restrict__ bias,
                                float* __restrict__ xnext,
                                long long n) {
    long long t = (long long)blockIdx.x * blockDim.x + threadIdx.x;
    if (t >= n) return;
    float v = acc[t] + bias[t % DOUT];
    xnext[t] = (v > 0.f) ? v : (__expf(v) - 1.f);
}

// ---------- host-side orchestration ----------
static inline long long cdiv_ll(long long a, long long b) { return (a + b - 1) / b; }

extern "C" void kernel_launch(void* const* d_in, const int* in_sizes, int n_in,
                              void* d_out, int out_size, void* d_ws, size_t ws_size,
                              hipStream_t stream) {
    const float* x0  = (const float*)d_in[0];
    const int*   ei  = (const int*)d_in[1];

    const int F = in_sizes[2] / DOUT;        // Wl0 is [F, 80]
    const int N = in_sizes[0] / F;
    const int E = in_sizes[1] / 2;
    const int* srcA = ei;
    const int* dstA = ei + E;
    const long long Etot = (long long)E + N;

    // workspace carve-up (256B aligned)
    char* ws = (char*)d_ws;
    size_t off = 0;
    auto carve = [&](size_t bytes) -> char* {
        char* p = ws + off;
        off += (bytes + 255) & ~(size_t)255;
        return p;
    };
    float*    xl    = (float*)carve((size_t)N * DOUT * sizeof(float));
    float*    xr    = (float*)carve((size_t)N * DOUT * sizeof(float));
    float*    ebuf  = (float*)carve((size_t)Etot * HEADS * sizeof(float));
    unsigned* nmax  = (unsigned*)carve((size_t)N * HEADS * sizeof(unsigned));
    float*    denom = (float*)carve((size_t)N * HEADS * sizeof(float));
    float*    acc   = (float*)carve((size_t)N * DOUT * sizeof(float));
    float*    xbuf  = (float*)carve((size_t)N * DOUT * sizeof(float));
    (void)ws_size;

    const int BLK = 256;                     // 8 waves per block
    const float* xcur = x0;
    int K = F;

    for (int l = 0; l < 3; ++l) {
        const float* Wl   = (const float*)d_in[2 + 4 * l];
        const float* Wr   = (const float*)d_in[3 + 4 * l];
        const float* att  = (const float*)d_in[4 + 4 * l];
        const float* bias = (const float*)d_in[5 + 4 * l];

        // 1) fused GEMMs: xl = x@Wl, xr = x@Wr (one wave per 16-row strip)
        {
            long long waves = cdiv_ll(N, 16);
            int grid = (int)cdiv_ll(waves, BLK / 32);
            size_t shmem = 2 * (size_t)K * DOUT * sizeof(float);
            if (K == 128)
                gemm2_wmma<128><<<grid, BLK, shmem, stream>>>(xcur, Wl, Wr, xl, xr, N, K);
            else if (K == DOUT)
                gemm2_wmma<DOUT><<<grid, BLK, shmem, stream>>>(xcur, Wl, Wr, xl, xr, N, K);
            else
                gemm2_wmma<0><<<grid, BLK, shmem, stream>>>(xcur, Wl, Wr, xl, xr, N, K);
        }

        // 2) init segment-max (ordered -inf == 0), denom, accumulator
        {
            long long nmx = (long long)N * HEADS;
            fill_u32<<<(int)cdiv_ll(nmx, BLK), BLK, 0, stream>>>(nmax, 0u, nmx);
            fill_u32<<<(int)cdiv_ll(nmx, BLK), BLK, 0, stream>>>((unsigned*)denom, 0u, nmx);
            long long na = (long long)N * DOUT;
            fill_u32<<<(int)cdiv_ll(na, BLK), BLK, 0, stream>>>((unsigned*)acc, 0u, na);
        }

        // 3) e + segment max
        {
            long long tot = Etot * HEADS;
            edge_e_kernel<<<(int)cdiv_ll(tot, BLK), BLK, 0, stream>>>(
                xl, xr, srcA, dstA, att, ebuf, nmax, E, N);
        }
        // 4) p + segment sum
        {
            long long tot = Etot * HEADS;
            edge_p_kernel<<<(int)cdiv_ll(tot, BLK), BLK, 0, stream>>>(
                ebuf, nmax, denom, dstA, E, N);
        }
        // 5) scatter alpha * xl[src]
        {
            long long tot = Etot * DOUT;
            edge_scatter_kernel<<<(int)cdiv_ll(tot, BLK), BLK, 0, stream>>>(
                xl, ebuf, denom, srcA, dstA, acc, E, N);
        }
        // 6) elu(acc + bias) -> next activation (last layer -> d_out)
        {
            long long na = (long long)N * DOUT;
            float* dst = (l == 2) ? (float*)d_out : xbuf;
            finalize_kernel<<<(int)cdiv_ll(na, BLK), BLK, 0, stream>>>(
                acc, bias, dst, na);
            xcur = dst;
            K = DOUT;
        }
    }
}